// IAFF_27917287424026
// MI455X (gfx1250) — compile-verified
//
#include <hip/hip_runtime.h>
#include <math.h>

#define NPTS   40000
#define NEDGE  1000000
#define KK     4
#define CELLS  16
#define BN_EPS 1e-5f
#define MAXKC  1024          // 16 * max(cin)=64

typedef __attribute__((ext_vector_type(2))) float v2f;
typedef __attribute__((ext_vector_type(8))) float v8f;

// ---------------- utility: zero a float4 region ----------------
__global__ void zero4_kernel(float4* p, int n4) {
    int i = blockIdx.x * blockDim.x + threadIdx.x;
    if (i < n4) p[i] = make_float4(0.f, 0.f, 0.f, 0.f);
}

// ---------------- concat xa = [x | y] ----------------
__global__ void concat_kernel(const float* __restrict__ x,
                              const float* __restrict__ y,
                              float* __restrict__ xa) {
    int i = blockIdx.x * blockDim.x + threadIdx.x;
    if (i >= NPTS * 64) return;
    int n = i >> 6, c = i & 63;
    xa[i] = (c < 32) ? x[n * 32 + c] : y[n * 32 + c - 32];
}

// ---------------- per-edge scatter: one wave per edge, lanes split channels ----------------
__global__ void scatter_kernel(const float* __restrict__ feat,
                               const int*  __restrict__ senders,
                               const int*  __restrict__ receivers,
                               const float* __restrict__ rel_pos,
                               const int*  __restrict__ wsp,
                               float* __restrict__ S, int cin) {
    int tid   = blockIdx.x * blockDim.x + threadIdx.x;
    int wave  = tid >> 5;
    int lane  = tid & 31;
    int nwave = (gridDim.x * blockDim.x) >> 5;
    float inv_ws = 1.0f / (float)wsp[0];

    for (int e = wave; e < NEDGE; e += nwave) {
        float2 rp = ((const float2*)rel_pos)[e];
        float ux = rp.x * inv_ws;
        float uy = rp.y * inv_ws;
        float t  = fmaxf(1.0f - (ux * ux + uy * uy), 0.0f);
        float win = t * t * t;
        if (win <= 0.0f) continue;  // uniform across wave (same edge)

        float gx = fminf(fmaxf((ux + 1.0f) * 0.5f * (float)(KK - 1), 0.0f), (float)(KK - 1));
        float gy = fminf(fmaxf((uy + 1.0f) * 0.5f * (float)(KK - 1), 0.0f), (float)(KK - 1));
        int ix = min((int)gx, KK - 2);   // gx >= 0 so trunc == floor
        int iy = min((int)gy, KK - 2);
        float fx = gx - (float)ix;
        float fy = gy - (float)iy;
        float w00 = (1.0f - fx) * (1.0f - fy) * win;
        float w01 = (1.0f - fx) * fy          * win;
        float w10 = fx          * (1.0f - fy) * win;
        float w11 = fx          * fy          * win;

        int s = senders[e], r = receivers[e];
        const float* f = feat + (size_t)s * cin;
        float* base = S + (size_t)r * CELLS * cin;
        int c00 = ( ix      * KK + iy    ) * cin;
        int c01 = ( ix      * KK + iy + 1) * cin;
        int c10 = ((ix + 1) * KK + iy    ) * cin;
        int c11 = ((ix + 1) * KK + iy + 1) * cin;

        for (int c = lane; c < cin; c += 32) {
            float fv = f[c];
            atomicAdd(base + c00 + c, w00 * fv);
            atomicAdd(base + c01 + c, w01 * fv);
            atomicAdd(base + c10 + c, w10 * fv);
            atomicAdd(base + c11 + c, w11 * fv);
        }
    }
}

// ---------------- WMMA GEMM with LDS-staged B ----------------
// out[N,cout] = S[N,KC] @ W[KC,cout], scaled by a[n]; fused BN partial sums.
// Each 256-thread block: one 16-col N-tile, 8 consecutive 16-row M-tiles.
// B tile (KC x 16) staged once into LDS, packed as [KC/2][16][2] so each
// lane's (k, k+1) pair for its column is a single aligned ds_load_b64.
__global__ void __launch_bounds__(256)
cconv_gemm_kernel(const float* __restrict__ S,
                  const float* __restrict__ W,
                  const float* __restrict__ avec,
                  float* __restrict__ out,
                  float* __restrict__ chsum,
                  float* __restrict__ chsq,
                  int KC, int cout) {
    __shared__ float bsh[MAXKC * 16];        // 64 KB (of 320 KB/WGP)

    int ntilesN = cout >> 4;
    int bm = blockIdx.x / ntilesN;           // M-block (8 M-tiles)
    int nt = blockIdx.x % ntilesN;           // N-tile
    int n0 = nt * 16;

    // ---- stage B tile: bsh[(k>>1)*32 + col*2 + (k&1)] = W[k][n0+col] ----
    for (int i = threadIdx.x; i < KC * 16; i += 256) {
        int k = i >> 4, c = i & 15;
        bsh[(k >> 1) * 32 + c * 2 + (k & 1)] = W[(size_t)k * cout + n0 + c];
    }
    __syncthreads();

    int wid  = threadIdx.x >> 5;
    int lane = threadIdx.x & 31;
    int mtile = bm * 8 + wid;
    if (mtile >= NPTS / 16) return;          // wave-uniform exit (after barrier)

    int m0 = mtile * 16;
    int half = lane >> 4;                    // 0: K=k0,k0+1 ; 1: K=k0+2,k0+3
    int lm   = lane & 15;

    const float* srow = S + (size_t)(m0 + lm) * KC + half * 2;   // A row
    const float* brow = bsh + half * 32 + lm * 2;                // B pair base

    v8f acc = {};
    #pragma unroll 4
    for (int k0 = 0; k0 < KC; k0 += 4) {
        float2 av = *(const float2*)(srow + k0);                 // global, 8B aligned
        float2 bv = *(const float2*)(brow + (k0 >> 1) * 32);     // LDS ds_load_b64
        v2f a = {av.x, av.y};
        v2f b = {bv.x, bv.y};
        acc = __builtin_amdgcn_wmma_f32_16x16x4_f32(
            /*neg_a=*/false, a, /*neg_b=*/false, b,
            /*c_mod=*/(short)0, acc, /*reuse_a=*/false, /*reuse_b=*/false);
    }

    // Epilogue: scale by a[row], store, accumulate per-channel BN sums.
    // C layout: VGPR r holds M = m0 + 8*half + r, N = n0 + lm.
    int col = n0 + lm;
    float ssum = 0.0f, ssq = 0.0f;
    #pragma unroll
    for (int r = 0; r < 8; ++r) {
        int row = m0 + half * 8 + r;
        float v = acc[r] * avec[row];
        out[(size_t)row * cout + col] = v;
        ssum += v;
        ssq  += v * v;
    }
    atomicAdd(&chsum[col], ssum);
    atomicAdd(&chsq[col],  ssq);
}

// ---------------- fold BN sums into per-channel scale/shift ----------------
__global__ void bnprep_kernel(const float* __restrict__ chsum,
                              const float* __restrict__ chsq,
                              const float* __restrict__ g,
                              const float* __restrict__ b,
                              float* __restrict__ scale,
                              float* __restrict__ shift, int cout) {
    int c = threadIdx.x;
    if (c >= cout) return;
    float mean = chsum[c] * (1.0f / (float)NPTS);
    float var  = chsq[c]  * (1.0f / (float)NPTS) - mean * mean;
    float inv  = rsqrtf(var + BN_EPS);
    float sc   = g[c] * inv;
    scale[c] = sc;
    shift[c] = b[c] - mean * sc;
}

// ---------------- BN + ReLU ----------------
__global__ void bn_relu_kernel(const float* __restrict__ t,
                               const float* __restrict__ scale,
                               const float* __restrict__ shift,
                               float* __restrict__ h, int cout) {
    int i = blockIdx.x * blockDim.x + threadIdx.x;
    if (i >= NPTS * cout) return;
    int c = i & (cout - 1);   // cout is a power of two
    h[i] = fmaxf(t[i] * scale[c] + shift[c], 0.0f);
}

// ---------------- BN + sigmoid gate blend: out = 2x*w + 2y*(1-w) ----------------
__global__ void bn_sig_combine_kernel(const float* __restrict__ t,
                                      const float* __restrict__ scale,
                                      const float* __restrict__ shift,
                                      const float* __restrict__ x,
                                      const float* __restrict__ y,
                                      float* __restrict__ out) {
    int i = blockIdx.x * blockDim.x + threadIdx.x;
    if (i >= NPTS * 32) return;
    int c = i & 31;
    float z = t[i] * scale[c] + shift[c];
    float w = 1.0f / (1.0f + __expf(-z));
    out[i] = 2.0f * x[i] * w + 2.0f * y[i] * (1.0f - w);
}

extern "C" void kernel_launch(void* const* d_in, const int* in_sizes, int n_in,
                              void* d_out, int out_size, void* d_ws, size_t ws_size,
                              hipStream_t stream) {
    const float* x        = (const float*)d_in[0];
    const float* y        = (const float*)d_in[1];
    const int*   senders  = (const int*)  d_in[2];
    const int*   receivers= (const int*)  d_in[3];
    const float* rel_pos  = (const float*)d_in[4];
    const int*   wsp      = (const int*)  d_in[5];
    const float* avec     = (const float*)d_in[6];
    const float* W1       = (const float*)d_in[7];
    const float* W2       = (const float*)d_in[8];
    const float* W3       = (const float*)d_in[9];
    const float* W4       = (const float*)d_in[10];
    const float* g1 = (const float*)d_in[11]; const float* b1 = (const float*)d_in[12];
    const float* g2 = (const float*)d_in[13]; const float* b2 = (const float*)d_in[14];
    const float* g3 = (const float*)d_in[15]; const float* b3 = (const float*)d_in[16];
    const float* g4 = (const float*)d_in[17]; const float* b4 = (const float*)d_in[18];
    float* outp = (float*)d_out;

    // workspace layout (floats)
    float* ws    = (float*)d_ws;
    float* xa    = ws;                           // N*64
    float* S     = xa + (size_t)NPTS * 64;       // N*16*64 (reused per layer)
    float* traw  = S  + (size_t)NPTS * 1024;     // N*64 raw cconv output
    float* hbuf  = traw + (size_t)NPTS * 64;     // N*64 activated features
    float* xo    = hbuf + (size_t)NPTS * 64;     // N*32
    float* stats = xo + (size_t)NPTS * 32;       // 256 floats
    float* chsum = stats;        // 64
    float* chsq  = stats + 64;   // 64
    float* scale = stats + 128;  // 64
    float* shift = stats + 192;  // 64

    auto run_layer = [&](const float* feat, int cin, const float* W, int cout,
                         const float* g, const float* b, float* raw) {
        int KC = CELLS * cin;
        int n4 = NPTS * CELLS * cin / 4;
        zero4_kernel<<<(n4 + 255) / 256, 256, 0, stream>>>((float4*)S, n4);
        zero4_kernel<<<1, 64, 0, stream>>>((float4*)stats, 32);   // zero chsum+chsq
        scatter_kernel<<<4096, 256, 0, stream>>>(feat, senders, receivers, rel_pos, wsp, S, cin);
        int mblocks = (NPTS / 16 + 7) / 8;            // 313
        int blocks  = mblocks * (cout / 16);
        cconv_gemm_kernel<<<blocks, 256, 0, stream>>>(S, W, avec, raw, chsum, chsq, KC, cout);
        bnprep_kernel<<<1, 64, 0, stream>>>(chsum, chsq, g, b, scale, shift, cout);
    };

    // xa = concat(x, y)
    concat_kernel<<<(NPTS * 64 + 255) / 256, 256, 0, stream>>>(x, y, xa);

    // layer 1: cconv(xa, W1) -> BN -> relu -> hbuf
    run_layer(xa, 64, W1, 64, g1, b1, traw);
    bn_relu_kernel<<<(NPTS * 64 + 255) / 256, 256, 0, stream>>>(traw, scale, shift, hbuf, 64);

    // layer 2: cconv(hbuf, W2) -> BN -> sigmoid gate -> xo
    run_layer(hbuf, 64, W2, 32, g2, b2, traw);
    bn_sig_combine_kernel<<<(NPTS * 32 + 255) / 256, 256, 0, stream>>>(traw, scale, shift, x, y, xo);

    // layer 3: cconv(xo, W3) -> BN -> relu -> hbuf
    run_layer(xo, 32, W3, 64, g3, b3, traw);
    bn_relu_kernel<<<(NPTS * 64 + 255) / 256, 256, 0, stream>>>(traw, scale, shift, hbuf, 64);

    // layer 4: cconv(hbuf, W4) -> BN -> sigmoid gate -> out
    run_layer(hbuf, 64, W4, 32, g4, b4, traw);
    bn_sig_combine_kernel<<<(NPTS * 32 + 255) / 256, 256, 0, stream>>>(traw, scale, shift, x, y, outp);
}